// PathSelector_32366873542911
// MI455X (gfx1250) — compile-verified
//
#include <hip/hip_runtime.h>
#include <hip/hip_bf16.h>

// Problem constants from the reference
#define BB 32
#define NN 64
#define CC 8
#define HH 64
#define PP 16
#define LL 8
#define EE 7           // L-1 edges per path
#define K2 128         // 2*H = GEMM K and N (W1 is 128x128)
#define STRIDE 132     // LDS row stride (pad 128 -> 132: conflict-free column reads)

typedef __attribute__((ext_vector_type(2))) float v2f;
typedef __attribute__((ext_vector_type(8))) float v8f;

__global__ __launch_bounds__(256) void pathsel_fused(
    const float* __restrict__ ef,            // (B,N,N,C,H)
    const float* __restrict__ gemb,          // (B,H)
    const int*   __restrict__ scom,          // (B,)
    const int*   __restrict__ paths,         // (B,P,L)
    const int*   __restrict__ plen,          // (B,P)
    const unsigned char* __restrict__ pmask, // (B,P) bool
    const float* __restrict__ W1,            // (128,128) row-major (k,n)
    const float* __restrict__ b1,            // (128,)
    const float* __restrict__ W2,            // (128,1)
    const float* __restrict__ b2,            // (1,)
    float* __restrict__ out)                 // [512 probs | 512 logp | 32 entropy]
{
    __shared__ float inp_s[PP * STRIDE];     // 16 rows x 128 cols (padded)
    __shared__ float score_s[PP];

    const int tid  = threadIdx.x;
    const int wave = tid >> 5;
    const int lane = tid & 31;
    const int b    = blockIdx.x;

    const int sc = scom[b];

    // ---------------- Phase 1: gather + masked mean + concat into LDS -------
    // wave w handles paths p = w and w+8; lanes cover h = lane and lane+32.
    for (int p = wave; p < PP; p += 8) {
        int len  = plen[b * PP + p];
        int ecnt = len - 1;
        if (ecnt < 0) ecnt = 0;
        if (ecnt > EE) ecnt = EE;
        int denom = ecnt > 0 ? ecnt : 1;
        float invd = 1.0f / (float)denom;

        const int* prow = paths + (size_t)(b * PP + p) * LL;

        for (int hh = 0; hh < HH; hh += 32) {
            int h = lane + hh;
            float acc = 0.0f;
            #pragma unroll
            for (int e = 0; e < EE; ++e) {
                if (e < ecnt) {      // uniform per wave (p is wave-uniform)
                    int u = prow[e];
                    int v = prow[e + 1];
                    size_t idx = ((((size_t)b * NN + u) * NN + v) * CC + sc) * (size_t)HH + h;
                    acc += ef[idx];
                }
            }
            inp_s[p * STRIDE + h]      = acc * invd;          // path_feat
            inp_s[p * STRIDE + HH + h] = gemb[b * HH + h];    // graph embedding
        }
    }
    if (tid < PP) score_s[tid] = b2[0];
    __syncthreads();

    // ---------------- Phase 2: f32 WMMA GEMM (16x128 @ 128x128) -------------
    // wave w owns N-tile n0 = 16*w. A from LDS, B from global W1.
    {
        const int n0 = wave * 16;
        const int nl = lane & 15;        // N within tile (B/C/D lane mapping)
        const int m  = lane & 15;        // M for A fragment
        const int hi = lane >> 4;
        const int kb = hi * 2;           // K sub-offset per lane-half

        v8f c = {0.f, 0.f, 0.f, 0.f, 0.f, 0.f, 0.f, 0.f};

        #pragma unroll
        for (int k0 = 0; k0 < K2; k0 += 4) {
            v2f a, bm;
            // A 16x4: VGPR0 = K=k0+kb, VGPR1 = K=k0+kb+1 (lanes 0-15: kb=0; 16-31: kb=2)
            a.x = inp_s[m * STRIDE + k0 + kb];
            a.y = inp_s[m * STRIDE + k0 + kb + 1];
            // B 4x16: same K striping, N = lane&15 within one VGPR
            bm.x = W1[(size_t)(k0 + kb)     * K2 + n0 + nl];
            bm.y = W1[(size_t)(k0 + kb + 1) * K2 + n0 + nl];
            c = __builtin_amdgcn_wmma_f32_16x16x4_f32(
                    /*neg_a=*/false, a, /*neg_b=*/false, bm,
                    /*c_mod=*/(short)0, c, /*reuse_a=*/false, /*reuse_b=*/false);
        }

        // bias + ReLU + W2 dot, then reduce over N (16 lanes per half) and
        // accumulate into per-path scores. D layout: VGPR r -> M = r + 8*hi.
        float b1n = b1[n0 + nl];
        float w2n = W2[n0 + nl];
        #pragma unroll
        for (int r = 0; r < 8; ++r) {
            float part = fmaxf(c[r] + b1n, 0.0f) * w2n;
            part += __shfl_xor(part, 1, 32);
            part += __shfl_xor(part, 2, 32);
            part += __shfl_xor(part, 4, 32);
            part += __shfl_xor(part, 8, 32);
            if (nl == 0) atomicAdd(&score_s[hi * 8 + r], part);  // ds_add_f32
        }
    }
    __syncthreads();

    // ---------------- Phase 3: masked softmax / log-softmax / entropy -------
    if (tid == 0) {
        const float NEG_INF = -__builtin_huge_valf();
        float s[PP], ex[PP];
        float mx = NEG_INF;
        #pragma unroll
        for (int p = 0; p < PP; ++p) {
            bool mk = pmask[b * PP + p] != 0;
            s[p] = mk ? score_s[p] : NEG_INF;
            mx = fmaxf(mx, s[p]);
        }
        float sum = 0.0f;
        #pragma unroll
        for (int p = 0; p < PP; ++p) { ex[p] = __expf(s[p] - mx); sum += ex[p]; }
        float inv = 1.0f / sum;
        float lse = __logf(sum);
        float ent = 0.0f;
        #pragma unroll
        for (int p = 0; p < PP; ++p) {
            float prob = ex[p] * inv;
            float logp = (s[p] - mx) - lse;       // -inf where masked out
            bool  mk   = pmask[b * PP + p] != 0;
            out[b * PP + p]            = prob;    // action_probs
            out[BB * PP + b * PP + p]  = logp;    // log_probs
            ent -= prob * (mk ? logp : 0.0f);
        }
        out[2 * BB * PP + b] = ent;               // entropy
    }
}

extern "C" void kernel_launch(void* const* d_in, const int* in_sizes, int n_in,
                              void* d_out, int out_size, void* d_ws, size_t ws_size,
                              hipStream_t stream) {
    (void)in_sizes; (void)n_in; (void)out_size; (void)d_ws; (void)ws_size;
    const float*         ef    = (const float*)d_in[0];
    const float*         gemb  = (const float*)d_in[1];
    const int*           scomm = (const int*)d_in[2];
    const int*           paths = (const int*)d_in[3];
    const int*           plen  = (const int*)d_in[4];
    const unsigned char* pmask = (const unsigned char*)d_in[5];
    const float*         W1    = (const float*)d_in[6];
    const float*         b1v   = (const float*)d_in[7];
    const float*         W2    = (const float*)d_in[8];
    const float*         b2v   = (const float*)d_in[9];
    float*               out   = (float*)d_out;

    pathsel_fused<<<BB, 256, 0, stream>>>(ef, gemb, scomm, paths, plen, pmask,
                                          W1, b1v, W2, b2v, out);
}